// LaplacianPosEnc_41540923687646
// MI455X (gfx1250) — compile-verified
//
#include <hip/hip_runtime.h>
#include <hip/hip_bf16.h>

// ---------------------------------------------------------------------------
// Laplacian positional encoding, CDNA5 (gfx1250), wave32.
//   M = 2I - L = I + D^{-1/2} A D^{-1/2}   (spectrum in [0,2])
//   * A staged into LDS by the Tensor Data Mover (tensor_load_to_lds) with
//     hardware pad insertion (1 DWORD per 32) -> conflict-free 165-stride.
//   * Block subspace iteration (block = 16) on M; M*V, Gram, whitening and
//     Rayleigh-Ritz all run on V_WMMA_F32_16X16X4_F32.
//   * Orthonormalization = Cholesky whitening (WMMA Gram -> serial chol ->
//     column-parallel L^{-T} -> WMMA apply).
//   * Final 16x16 Jacobi + sort, drop the trivial pair, emit 8 vectors.
// ---------------------------------------------------------------------------

#define NN     160          // matrix size
#define NB     16           // subspace block width (WMMA N)
#define KOUT   8            // eigenvectors emitted
#define NITER  64           // subspace iterations
#define NTHR   320          // 10 waves of 32
#define NWAVE  10
#define MSZ    26400        // padded M size in floats (>= MOFF(159,159)+1)

// TDM pad layout: 1 pad DWORD after every 32 data DWORDs
// element (r,c) lives at DWORD offset r*165 + c + (c>>5); 165 % 64 = 37,
// gcd(37,64)=1 -> 16-lane row accesses are bank-conflict-free.
#define MOFF(r, c) ((r) * 165 + (c) + ((c) >> 5))

typedef unsigned int u32;
typedef __attribute__((ext_vector_type(2))) float v2f;
typedef __attribute__((ext_vector_type(8))) float v8f;
typedef __attribute__((ext_vector_type(4))) unsigned int v4u;
typedef __attribute__((ext_vector_type(8))) int v8i;
typedef __attribute__((ext_vector_type(4))) int v4i;

#define WMMA_F32(a, b, c) \
    __builtin_amdgcn_wmma_f32_16x16x4_f32(false, (a), false, (b), (short)0, (c), false, false)

// One TDM descriptor: 2-D tile 160x160 f32, global -> LDS with pad insertion.
__device__ __forceinline__ void tdm_load_A(const float* Ab, float* Ms) {
    const u32 lds = (u32)(uintptr_t)Ms;            // flat[31:0] == LDS offset
    const unsigned long long ga = (unsigned long long)(uintptr_t)Ab;

    v4u g0;
    g0.x = 1u;                                     // count=1 (valid user D#)
    g0.y = lds;                                    // lds_addr (bytes)
    g0.z = (u32)(ga & 0xFFFFFFFFu);                // global_addr[31:0]
    g0.w = (u32)((ga >> 32) & 0x01FFFFFFu)         // global_addr[56:32]
         | (2u << 30);                             // type = 2 ("image")

    v8i g1;
    g1[0] = (int)((2u << 16)                       // data_size = 4 bytes
                | (1u << 20)                       // pad_enable
                | (4u << 22)                       // pad_interval: 32 DWORDs
                | (0u << 25));                     // pad_amount: 1 DWORD
    g1[1] = (int)(160u << 16);                     // tensor_dim0 = 160
    g1[2] = (int)(160u << 16);                     // tensor_dim1 = 160
    g1[3] = (int)(160u << 16);                     // tile_dim0 = 160
    g1[4] = (int)160;                              // tile_dim1 = 160 (dim2=0)
    g1[5] = (int)160;                              // tensor_dim0_stride = 160
    g1[6] = 0;
    g1[7] = 0;

    v4i gz = {0, 0, 0, 0};
#if __clang_major__ >= 23
    v8i gz8 = {0, 0, 0, 0, 0, 0, 0, 0};
    __builtin_amdgcn_tensor_load_to_lds(g0, g1, gz, gz, gz8, 0);
#else
    __builtin_amdgcn_tensor_load_to_lds(g0, g1, gz, gz, 0);
#endif
    __builtin_amdgcn_s_wait_tensorcnt(0);
}

// W(160x16) = M(160x160, padded) * V(160x16). Wave w owns rows [16w,16w+16).
// Two accumulators break the dependent WMMA chain.
__device__ __forceinline__ void gemm_MV(const float* Ms, const float* Vsrc,
                                        float* Wdst, int tid) {
    const int lane = tid & 31;
    const int w    = tid >> 5;
    const int hf   = lane >> 4;
    const int l    = lane & 15;
    const int arow = (w * 16 + l) * 165;

    v8f acc0 = {0.f, 0.f, 0.f, 0.f, 0.f, 0.f, 0.f, 0.f};
    v8f acc1 = acc0;
#pragma unroll 2
    for (int k0 = 0; k0 < NN; k0 += 8) {
        const int ka = k0 + hf * 2;
        const int kb = ka + 4;
        v2f a0;  a0.x = Ms[arow + ka + (ka >> 5)];
                 a0.y = Ms[arow + ka + (ka >> 5) + 1];
        v2f b0;  b0.x = Vsrc[ka * NB + l];
                 b0.y = Vsrc[(ka + 1) * NB + l];
        v2f a1;  a1.x = Ms[arow + kb + (kb >> 5)];
                 a1.y = Ms[arow + kb + (kb >> 5) + 1];
        v2f b1;  b1.x = Vsrc[kb * NB + l];
                 b1.y = Vsrc[(kb + 1) * NB + l];
        acc0 = WMMA_F32(a0, b0, acc0);
        acc1 = WMMA_F32(a1, b1, acc1);
    }
    const int row0 = w * 16 + hf * 8;
#pragma unroll
    for (int r = 0; r < 8; ++r)
        Wdst[(row0 + r) * NB + l] = acc0[r] + acc1[r];
    __syncthreads();
}

// Gout(16x16) = X^T(16x160) * Y(160x16) via per-wave K-slices + LDS reduce.
// For X^T as the A operand the lane layout equals the B layout, so X==Y
// needs only one set of loads. Also zeroes Bl for the following cholinv.
__device__ __forceinline__ void gram16(const float* X, const float* Y,
                                       float* Gpart, float* Gout, float* Bl,
                                       int tid) {
    const int lane  = tid & 31;
    const int w     = tid >> 5;
    const int hf    = lane >> 4;
    const int l     = lane & 15;
    const int kbase = w * 16;

    v8f acc = {0.f, 0.f, 0.f, 0.f, 0.f, 0.f, 0.f, 0.f};
#pragma unroll
    for (int s = 0; s < 4; ++s) {
        const int ka = kbase + s * 4 + hf * 2;
        v2f a;  a.x  = X[ka * NB + l];
                a.y  = X[(ka + 1) * NB + l];
        v2f bv; bv.x = Y[ka * NB + l];
                bv.y = Y[(ka + 1) * NB + l];
        acc = WMMA_F32(a, bv, acc);
    }
    const int row0 = hf * 8;
#pragma unroll
    for (int r = 0; r < 8; ++r)
        Gpart[w * 256 + (row0 + r) * 16 + l] = acc[r];
    __syncthreads();

    if (tid < 256) {
        float s = 0.0f;
#pragma unroll
        for (int w2 = 0; w2 < NWAVE; ++w2) s += Gpart[w2 * 256 + tid];
        Gout[tid] = s;
        Bl[tid]   = 0.0f;
    }
    __syncthreads();
}

// Gs (SPD 16x16) -> Bl = L^{-T} row-major (upper triangular), Gs <- L in place.
__device__ __forceinline__ void cholinv(float* Gs, float* Bl, int tid) {
    if (tid == 0) {
        for (int k = 0; k < 16; ++k) {
            float s = Gs[k * 16 + k];
            for (int m = 0; m < k; ++m) { float v = Gs[k * 16 + m]; s -= v * v; }
            float dk = sqrtf(fmaxf(s, 1e-20f));
            Gs[k * 16 + k] = dk;
            float inv = 1.0f / dk;
            for (int i = k + 1; i < 16; ++i) {
                float t = Gs[i * 16 + k];
                for (int m = 0; m < k; ++m) t -= Gs[i * 16 + m] * Gs[k * 16 + m];
                Gs[i * 16 + k] = t * inv;
            }
        }
    }
    __syncthreads();
    // column j of L^{-1} by forward substitution; Bl[j][i] = (L^{-1})[i][j]
    if (tid < 16) {
        const int j = tid;
        Bl[j * 16 + j] = 1.0f / Gs[j * 16 + j];
        for (int i = j + 1; i < 16; ++i) {
            float t = 0.0f;
            for (int m = j; m < i; ++m) t += Gs[i * 16 + m] * Bl[j * 16 + m];
            Bl[j * 16 + i] = -t / Gs[i * 16 + i];
        }
    }
    __syncthreads();
}

// Y(160x16) = X(160x16) * Bm(16x16). In-place X==Y is safe: each wave reads
// only the 16 rows it writes, and same-wave DS ops are in order.
__device__ __forceinline__ void gemm_nb(const float* X, const float* Bm,
                                        float* Y, int tid) {
    const int lane = tid & 31;
    const int w    = tid >> 5;
    const int hf   = lane >> 4;
    const int l    = lane & 15;
    const int arow = w * 16 + l;

    v8f acc = {0.f, 0.f, 0.f, 0.f, 0.f, 0.f, 0.f, 0.f};
#pragma unroll
    for (int s = 0; s < 4; ++s) {
        const int ka = s * 4 + hf * 2;
        v2f a;  a.x  = X[arow * NB + ka];
                a.y  = X[arow * NB + ka + 1];
        v2f bv; bv.x = Bm[ka * 16 + l];
                bv.y = Bm[(ka + 1) * 16 + l];
        acc = WMMA_F32(a, bv, acc);
    }
    const int row0 = w * 16 + hf * 8;
#pragma unroll
    for (int r = 0; r < 8; ++r)
        Y[(row0 + r) * NB + l] = acc[r];
    __syncthreads();
}

__global__ __launch_bounds__(NTHR)
void lappe_kernel(const float* __restrict__ A, float* __restrict__ out) {
    extern __shared__ float smem[];
    float* Ms    = smem;                   // MSZ      padded matrix M
    float* Vs    = Ms + MSZ;               // 160*16   subspace basis
    float* Ws    = Vs + NN * NB;           // 160*16   M*V workspace
    float* dinv  = Ws + NN * NB;           // 160      (d+eps)^-1/2
    float* Gpart = dinv + NN;              // 10*256   per-wave Gram partials
    float* Gs    = Gpart + NWAVE * 256;    // 16x16    Gram / chol factor
    float* Bl    = Gs + 256;               // 16x16    L^{-T} (row-major)
    float* Ts    = Bl + 256;               // 16x16    Rayleigh-Ritz matrix
    float* Us    = Ts + 256;               // 16x16    Jacobi rotations
    __shared__ int ord[NB];

    const int tid = threadIdx.x;
    const int b   = blockIdx.x;
    const float* Ab = A + (size_t)b * NN * NN;

    // ---- TDM: DMA A into padded LDS layout (wave 0 issues + waits) --------
    if (tid < 32) tdm_load_A(Ab, Ms);
    __syncthreads();

    // ---- degrees and D^{-1/2} ---------------------------------------------
    if (tid < NN) {
        float s = 0.0f;
        for (int c = 0; c < NN; ++c) s += Ms[MOFF(tid, c)];
        dinv[tid] = rsqrtf(s + 1e-8f);           // (d + EPS)^-0.5
    }
    __syncthreads();

    // ---- M = I + D^{-1/2} A D^{-1/2} in place -----------------------------
    for (int idx = tid; idx < NN * NN; idx += NTHR) {
        int r = idx / NN, c = idx - r * NN;
        float v = Ms[MOFF(r, c)] * dinv[r] * dinv[c];
        Ms[MOFF(r, c)] = v + ((r == c) ? 1.0f : 0.0f);
    }
    __syncthreads();

    // ---- init subspace: col0 = sqrt(d) (trivial evec), rest hashed --------
    if (tid < NN) {
        Vs[tid * NB + 0] = 1.0f / dinv[tid];     // sqrt(d + eps)
        for (int j = 1; j < NB; ++j) {
            unsigned h = (unsigned)(tid * NB + j) * 2654435761u + 0x9E3779B9u;
            h ^= h >> 13; h *= 0x5bd1e995u; h ^= h >> 15;
            Vs[tid * NB + j] = (float)(h & 0xFFFFu) * (1.0f / 65536.0f) - 0.5f;
        }
    }
    __syncthreads();

    // ---- initial whitening ------------------------------------------------
    gram16(Vs, Vs, Gpart, Gs, Bl, tid);
    cholinv(Gs, Bl, tid);
    gemm_nb(Vs, Bl, Vs, tid);

    // ---- subspace iteration: V <- (M V) L^{-T} ----------------------------
    for (int it = 0; it < NITER; ++it) {
        gemm_MV(Ms, Vs, Ws, tid);            // W = M V
        gram16(Ws, Ws, Gpart, Gs, Bl, tid);  // G = W^T W
        cholinv(Gs, Bl, tid);                // Bl = L^{-T}
        gemm_nb(Ws, Bl, Vs, tid);            // V = W L^{-T}
    }

    // ---- Rayleigh-Ritz: T = V^T (M V) -------------------------------------
    gemm_MV(Ms, Vs, Ws, tid);
    gram16(Vs, Ws, Gpart, Ts, Bl, tid);

    // ---- serial 16x16 Jacobi + sort (thread 0) ----------------------------
    if (tid == 0) {
        for (int i = 0; i < NB; ++i)
            for (int j = 0; j < NB; ++j)
                Us[i * 16 + j] = (i == j) ? 1.0f : 0.0f;
        for (int sweep = 0; sweep < 12; ++sweep)
            for (int p = 0; p < NB - 1; ++p)
                for (int q = p + 1; q < NB; ++q) {
                    float apq = Ts[p * 16 + q];
                    if (fabsf(apq) < 1e-12f) continue;
                    float app = Ts[p * 16 + p], aqq = Ts[q * 16 + q];
                    float th = 0.5f * (aqq - app) / apq;
                    float t  = 1.0f / (fabsf(th) + sqrtf(1.0f + th * th));
                    if (th < 0.0f) t = -t;
                    float c = rsqrtf(1.0f + t * t);
                    float s = t * c;
                    for (int k = 0; k < NB; ++k) {
                        float tp = Ts[p * 16 + k], tq = Ts[q * 16 + k];
                        Ts[p * 16 + k] = c * tp - s * tq;
                        Ts[q * 16 + k] = s * tp + c * tq;
                    }
                    for (int k = 0; k < NB; ++k) {
                        float tp = Ts[k * 16 + p], tq = Ts[k * 16 + q];
                        Ts[k * 16 + p] = c * tp - s * tq;
                        Ts[k * 16 + q] = s * tp + c * tq;
                        float up = Us[k * 16 + p], uq = Us[k * 16 + q];
                        Us[k * 16 + p] = c * up - s * uq;
                        Us[k * 16 + q] = s * up + c * uq;
                    }
                }
        // sort descending in M  <=>  ascending in L
        float vals[NB]; int idx[NB];
        for (int i = 0; i < NB; ++i) { vals[i] = Ts[i * 16 + i]; idx[i] = i; }
        for (int i = 0; i < NB; ++i) {
            int m = i;
            for (int j = i + 1; j < NB; ++j)
                if (vals[idx[j]] > vals[idx[m]]) m = j;
            int t = idx[i]; idx[i] = idx[m]; idx[m] = t;
        }
        for (int i = 0; i < NB; ++i) ord[i] = idx[i];
    }
    __syncthreads();

    // ---- emit Y = V * U[:, ord[1..8]]  as (B, N, 8) -----------------------
    if (tid < NN) {
        float* ob = out + ((size_t)b * NN + tid) * KOUT;
#pragma unroll
        for (int k = 0; k < KOUT; ++k) {
            int c = ord[k + 1];                 // skip trivial eigenvector
            float s = 0.0f;
            for (int j = 0; j < NB; ++j)
                s += Vs[tid * NB + j] * Us[j * 16 + c];
            ob[k] = s;
        }
    }
}

extern "C" void kernel_launch(void* const* d_in, const int* in_sizes, int n_in,
                              void* d_out, int out_size, void* d_ws, size_t ws_size,
                              hipStream_t stream) {
    (void)n_in; (void)out_size; (void)d_ws; (void)ws_size;
    const float* A = (const float*)d_in[0];
    float* out = (float*)d_out;
    const int B = in_sizes[0] / (NN * NN);
    const size_t shmem =
        (size_t)(MSZ + 2 * NN * NB + NN + NWAVE * 256 + 4 * 256) * sizeof(float);
    lappe_kernel<<<dim3(B), dim3(NTHR), shmem, stream>>>(A, out);
}